// GAugMAE_model_31018253811971
// MI455X (gfx1250) — compile-verified
//
#include <hip/hip_runtime.h>
#include <hip/hip_bf16.h>

// ---------------------------------------------------------------------------
// GAugMAE forward for MI455X (gfx1250, wave32, WMMA).
// fp32 end-to-end using V_WMMA_F32_16X16X4_F32 (preserves reference numerics
// through the round() threshold). N x N matrices (64 MB) are L2-resident
// (192 MB L2). Each wave register-blocks a 32x64 (or 32x16) C tile: per
// K-step of 4 it issues 8 WMMAs against 2 A-fragments x 4 B-fragments,
// cutting A traffic 4x and B loads to 1 b32 per WMMA.
// ---------------------------------------------------------------------------

typedef __attribute__((ext_vector_type(2))) float v2f;
typedef __attribute__((ext_vector_type(8))) float v8f;

#define NN   4096
#define FF   512
#define HH   256
#define DZ   64
#define CC   16
#define ALPHA 0.8f

// ---------------------------------------------------------------------------
// WMMA f32 GEMM: C[M,Ncol] = A[M,K] @ B  (B is [K,Ncol], or [Ncol,K] if TRANSB)
// One wave computes an (RT*16) x (CT*16) C tile, stepping K by 4 with
// V_WMMA_F32_16X16X4_F32.
//
// VGPR fragment layout (ISA 7.12.2, 32-bit A 16x4):
//   lanes 0-15 : row M=l, {K, K+1} in the two VGPRs
//   lanes 16-31: row M=l, {K+2, K+3}
// B (4x16) mirrored: lanes 0-15 col N=l rows {K,K+1}, lanes 16-31 rows {K+2,K+3}
// C/D (v8f): VGPR q holds M = q + 8*(lane/16), N = lane%16.
// ---------------------------------------------------------------------------
template<int RT, int CT, bool TRANSB, bool BIAS, bool RELU>
__global__ __launch_bounds__(32)
void wmma_gemm_f32(const float* __restrict__ A, const float* __restrict__ B,
                   const float* __restrict__ bias, float* __restrict__ C,
                   int K, int lda, int ldb, int ldc)
{
    const int lane = threadIdx.x & 31;
    const int half = lane >> 4;     // 0: {K,K+1}   1: {K+2,K+3}
    const int l    = lane & 15;
    const int row0 = blockIdx.y * (RT * 16);
    const int col0 = blockIdx.x * (CT * 16);

    v8f acc[RT][CT];
    #pragma unroll
    for (int r = 0; r < RT; ++r)
        #pragma unroll
        for (int c = 0; c < CT; ++c)
            acc[r][c] = 0.f;

    const float* arow[RT];
    #pragma unroll
    for (int r = 0; r < RT; ++r)
        arow[r] = A + (size_t)(row0 + r * 16 + l) * lda + 2 * half;

    const float* brow[CT];
    #pragma unroll
    for (int c = 0; c < CT; ++c)
        brow[c] = TRANSB ? (B + (size_t)(col0 + c * 16 + l) * ldb + 2 * half)
                         : (B + col0 + c * 16 + l);

    #pragma unroll 2
    for (int k = 0; k < K; k += 4) {
        v2f av[RT];
        #pragma unroll
        for (int r = 0; r < RT; ++r)
            av[r] = *(const v2f*)(arow[r] + k);

        v2f bv[CT];
        #pragma unroll
        for (int c = 0; c < CT; ++c) {
            if (TRANSB) {
                bv[c] = *(const v2f*)(brow[c] + k);
            } else {
                const size_t kk = (size_t)(k + 2 * half) * ldb;
                bv[c].x = brow[c][kk];
                bv[c].y = brow[c][kk + ldb];
            }
        }

        #pragma unroll
        for (int r = 0; r < RT; ++r)
            #pragma unroll
            for (int c = 0; c < CT; ++c)
                // 8 args: (neg_a, A, neg_b, B, c_mod, C, reuse_a, reuse_b)
                acc[r][c] = __builtin_amdgcn_wmma_f32_16x16x4_f32(
                    false, av[r], false, bv[c], (short)0, acc[r][c], false, false);
    }

    #pragma unroll
    for (int c = 0; c < CT; ++c) {
        const float bz = BIAS ? bias[col0 + c * 16 + l] : 0.f;
        #pragma unroll
        for (int r = 0; r < RT; ++r) {
            float* crow = C + (size_t)(row0 + r * 16 + 8 * half) * ldc
                            + col0 + c * 16 + l;
            #pragma unroll
            for (int q = 0; q < 8; ++q) {
                float v = acc[r][c][q] + bz;
                if (RELU) v = fmaxf(v, 0.f);
                crow[(size_t)q * ldc] = v;
            }
        }
    }
}

// ---------------------------------------------------------------------------
// Two-stage global max reduction (wave32 shuffles, 8 waves / block).
// ---------------------------------------------------------------------------
__global__ __launch_bounds__(256)
void kmax_reduce(const float* __restrict__ x, float* __restrict__ out, int n)
{
    float m = -3.402823466e38f;
    for (int i = blockIdx.x * blockDim.x + threadIdx.x; i < n;
         i += gridDim.x * blockDim.x)
        m = fmaxf(m, x[i]);
    #pragma unroll
    for (int off = 16; off; off >>= 1)
        m = fmaxf(m, __shfl_xor(m, off, 32));
    __shared__ float sm[8];
    if ((threadIdx.x & 31) == 0) sm[threadIdx.x >> 5] = m;
    __syncthreads();
    if (threadIdx.x == 0) {
        float t = sm[0];
        #pragma unroll
        for (int w = 1; w < 8; ++w) t = fmaxf(t, sm[w]);
        out[blockIdx.x] = t;
    }
}

// Resampled edge value (symmetric; diag forced to 1). jnp.round == rintf (RNE).
__device__ __forceinline__ float edge_val(float logit, float orig, float inv_max,
                                          int i, int j)
{
    if (i == j) return 1.0f;
    return rintf(ALPHA * logit * inv_max + (1.0f - ALPHA) * orig);
}

// ---------------------------------------------------------------------------
// Per-row degree -> dn[i] = rowsum^-0.5. One block per row.
// ---------------------------------------------------------------------------
__global__ __launch_bounds__(256)
void krow_dn(const float* __restrict__ logits, const float* __restrict__ adj_orig,
             const float* __restrict__ maxv, float* __restrict__ dn)
{
    const int i = blockIdx.x;
    const float inv = 1.0f / maxv[0];
    const size_t base = (size_t)i * NN;
    float s = 0.f;
    for (int j = threadIdx.x; j < NN; j += blockDim.x)
        s += edge_val(logits[base + j], adj_orig[base + j], inv, i, j);
    #pragma unroll
    for (int off = 16; off; off >>= 1)
        s += __shfl_xor(s, off, 32);
    __shared__ float sm[8];
    if ((threadIdx.x & 31) == 0) sm[threadIdx.x >> 5] = s;
    __syncthreads();
    if (threadIdx.x == 0) {
        float t = 0.f;
        #pragma unroll
        for (int w = 0; w < 8; ++w) t += sm[w];
        dn[i] = rsqrtf(t);
    }
}

// ---------------------------------------------------------------------------
// Materialize adj_new_normed[i,j] = a_ij * dn[i] * dn[j]  (64 MB, L2-resident).
// ---------------------------------------------------------------------------
__global__ __launch_bounds__(256)
void kbuild_adjn(const float* __restrict__ logits, const float* __restrict__ adj_orig,
                 const float* __restrict__ maxv, const float* __restrict__ dn,
                 float* __restrict__ adjn)
{
    const size_t idx = (size_t)blockIdx.x * blockDim.x + threadIdx.x;
    const int i = (int)(idx >> 12);      // / 4096
    const int j = (int)(idx & 4095);
    const float inv = 1.0f / maxv[0];
    const float a = edge_val(logits[idx], adj_orig[idx], inv, i, j);
    adjn[idx] = a * dn[i] * dn[j];
}

// ---------------------------------------------------------------------------
extern "C" void kernel_launch(void* const* d_in, const int* in_sizes, int n_in,
                              void* d_out, int out_size, void* d_ws, size_t ws_size,
                              hipStream_t stream)
{
    (void)in_sizes; (void)n_in; (void)out_size; (void)ws_size;

    const float* adj      = (const float*)d_in[0];
    const float* adj_orig = (const float*)d_in[1];
    const float* features = (const float*)d_in[2];
    const float* Wb       = (const float*)d_in[3];
    const float* Wm       = (const float*)d_in[4];
    const float* W0       = (const float*)d_in[5];
    const float* b0       = (const float*)d_in[6];
    const float* W1       = (const float*)d_in[7];
    const float* b1       = (const float*)d_in[8];
    const float* W2       = (const float*)d_in[9];
    const float* b2       = (const float*)d_in[10];

    float* nc_logits  = (float*)d_out;                       // [N, C]
    float* adj_logits = (float*)d_out + (size_t)NN * CC;     // [N, N]

    // Workspace carve-up (floats). ~82 MB total.
    float* ws = (float*)d_ws;
    size_t o = 0;
    float* tA   = ws + o; o += (size_t)NN * HH;   // [N,H] staging
    float* tB   = ws + o; o += (size_t)NN * HH;   // [N,H] staging
    float* tC   = ws + o; o += (size_t)NN * HH;   // [N,H] staging (also [N,DZ]/[N,C])
    float* mean = ws + o; o += (size_t)NN * DZ;   // [N,DZ]
    float* dn   = ws + o; o += NN;                // [N]
    float* pmax = ws + o; o += 4096;              // partial maxes
    float* maxv = ws + o; o += 64;                // global max (padded)
    float* adjn = ws + o;                         // [N,N] resampled normalized adj

    const dim3 blk(32);
    // wave tile = (RT*16) rows x (CT*16) cols
    auto grd = [](int m, int ncol, int rt, int ct) {
        return dim3(ncol / (16 * ct), m / (16 * rt));
    };

    // ---------------- ep_net (VGAE, gae=True) ----------------
    // tA = features @ Wb
    wmma_gemm_f32<2,4,false,false,false><<<grd(NN,HH,2,4), blk, 0, stream>>>(
        features, Wb, nullptr, tA, FF, FF, HH, HH);
    // tB = adj @ tA                                  (hidden, linear)
    wmma_gemm_f32<2,4,false,false,false><<<grd(NN,HH,2,4), blk, 0, stream>>>(
        adj, tA, nullptr, tB, NN, NN, HH, HH);
    // tC = tB @ Wm
    wmma_gemm_f32<2,4,false,false,false><<<grd(NN,DZ,2,4), blk, 0, stream>>>(
        tB, Wm, nullptr, tC, HH, HH, DZ, DZ);
    // mean = relu(adj @ tC)
    wmma_gemm_f32<2,4,false,false,true ><<<grd(NN,DZ,2,4), blk, 0, stream>>>(
        adj, tC, nullptr, mean, NN, NN, DZ, DZ);
    // adj_logits = mean @ mean.T
    wmma_gemm_f32<2,4,true ,false,false><<<grd(NN,NN,2,4), blk, 0, stream>>>(
        mean, mean, nullptr, adj_logits, DZ, DZ, DZ, NN);

    // ---------------- sample_adj_add_round + normalize ----------------
    kmax_reduce<<<4096, 256, 0, stream>>>(adj_logits, pmax, NN * NN);
    kmax_reduce<<<1,    256, 0, stream>>>(pmax, maxv, 4096);
    krow_dn<<<NN, 256, 0, stream>>>(adj_logits, adj_orig, maxv, dn);
    kbuild_adjn<<<(NN / 256) * NN, 256, 0, stream>>>(adj_logits, adj_orig, maxv, dn, adjn);

    // ---------------- nc_net (3-layer GCN) ----------------
    // tA = features @ W0 ; tB = relu(adjn @ tA + b0)
    wmma_gemm_f32<2,4,false,false,false><<<grd(NN,HH,2,4), blk, 0, stream>>>(
        features, W0, nullptr, tA, FF, FF, HH, HH);
    wmma_gemm_f32<2,4,false,true ,true ><<<grd(NN,HH,2,4), blk, 0, stream>>>(
        adjn, tA, b0, tB, NN, NN, HH, HH);
    // tC = tB @ W1 ; tA = relu(adjn @ tC + b1)
    wmma_gemm_f32<2,4,false,false,false><<<grd(NN,HH,2,4), blk, 0, stream>>>(
        tB, W1, nullptr, tC, HH, HH, HH, HH);
    wmma_gemm_f32<2,4,false,true ,true ><<<grd(NN,HH,2,4), blk, 0, stream>>>(
        adjn, tC, b1, tA, NN, NN, HH, HH);
    // tC = tA @ W2 ; nc_logits = adjn @ tC + b2   (Ncol = 16 -> CT = 1)
    wmma_gemm_f32<2,1,false,false,false><<<grd(NN,CC,2,1), blk, 0, stream>>>(
        tA, W2, nullptr, tC, HH, HH, CC, CC);
    wmma_gemm_f32<2,1,false,true ,false><<<grd(NN,CC,2,1), blk, 0, stream>>>(
        adjn, tC, b2, nc_logits, NN, NN, CC, CC);
}